// GumbelSampler_19791209300207
// MI455X (gfx1250) — compile-verified
//
#include <hip/hip_runtime.h>

typedef float v4f __attribute__((ext_vector_type(4)));

#define WAVES_PER_BLOCK 8
#define ROWS_PER_WAVE   4
#define ROWS_PER_BLOCK  (WAVES_PER_BLOCK * ROWS_PER_WAVE)   // 32
#define E               64
#define EPL             8    // elements per lane
#define KSEL            8

// ---- DPP16 helpers: butterfly within 8-lane groups, full-rate VALU, no LDS
#define DPP_XOR1  0xB1   // quad_perm(1,0,3,2)
#define DPP_XOR2  0x4E   // quad_perm(2,3,0,1)
#define DPP_HMIR  0x141  // row_half_mirror (pairs each quad with the other quad of its 8-group)

template <int CTRL>
__device__ __forceinline__ int dpp_i(int x) {
    return __builtin_amdgcn_update_dpp(0, x, CTRL, 0xF, 0xF, true);
}
template <int CTRL>
__device__ __forceinline__ float dpp_f(float x) {
    return __int_as_float(dpp_i<CTRL>(__float_as_int(x)));
}

__device__ __forceinline__ float group8_max(float x) {
    x = fmaxf(x, dpp_f<DPP_XOR1>(x));
    x = fmaxf(x, dpp_f<DPP_XOR2>(x));
    x = fmaxf(x, dpp_f<DPP_HMIR>(x));
    return x;
}
__device__ __forceinline__ float group8_sum(float x) {
    x += dpp_f<DPP_XOR1>(x);
    x += dpp_f<DPP_XOR2>(x);
    x += dpp_f<DPP_HMIR>(x);
    return x;
}
// argmax with lowest-index tie-break; after 3 steps all 8 lanes converge
template <int CTRL>
__device__ __forceinline__ void argmax_step(float& v, int& idx) {
    float ov = dpp_f<CTRL>(v);
    int   oi = dpp_i<CTRL>(idx);
    bool take = (ov > v) || ((ov == v) && (oi < idx));
    v   = take ? ov : v;
    idx = take ? oi : idx;
}

__global__ __launch_bounds__(256) void gumbel_relaxed_topk_kernel(
    const float* __restrict__ scores,
    const float* __restrict__ g,
    float* __restrict__ out,
    int nrows)
{
    // per wave: 4 rows x (64 scores + 64 gumbel) floats = 2 KB; block = 16 KB
    __shared__ __align__(16) float lds[WAVES_PER_BLOCK * 2 * ROWS_PER_WAVE * E];

    const int lane = threadIdx.x & 31;
    const int wave = threadIdx.x >> 5;
    const int gidx = lane >> 3;        // which of the 4 rows this lane serves
    const int s    = lane & 7;         // sub-lane within the 8-lane row group

    const long long waveRow0 =
        (long long)blockIdx.x * ROWS_PER_BLOCK + wave * ROWS_PER_WAVE;
    if (waveRow0 >= nrows) return;     // uniform per wave -> EXEC stays all-ones
    const long long row = waveRow0 + gidx;

    // ---- CDNA5 async DMA: stage 4 rows of scores (1KB) + g (1KB) into LDS.
    // Each b128 async op moves 512 B (32 lanes x 16 B); INST_OFFSET is added
    // to both the global and the LDS address, so two ops cover each 1 KB run.
    unsigned ldsS = (unsigned)(unsigned long long)(&lds[0])
                  + (unsigned)(wave * 2048 + lane * 16);
    unsigned ldsG = ldsS + 1024u;
    unsigned long long gaS =
        (unsigned long long)(scores + waveRow0 * E) + (unsigned long long)(lane * 16);
    unsigned long long gaG =
        (unsigned long long)(g + waveRow0 * E) + (unsigned long long)(lane * 16);
    asm volatile("global_load_async_to_lds_b128 %0, %1, off"
                 :: "v"(ldsS), "v"(gaS) : "memory");
    asm volatile("global_load_async_to_lds_b128 %0, %1, off offset:512"
                 :: "v"(ldsS), "v"(gaS) : "memory");
    asm volatile("global_load_async_to_lds_b128 %0, %1, off"
                 :: "v"(ldsG), "v"(gaG) : "memory");
    asm volatile("global_load_async_to_lds_b128 %0, %1, off offset:512"
                 :: "v"(ldsG), "v"(gaG) : "memory");
    asm volatile("s_wait_asynccnt 0" ::: "memory");

    const float* ws = &lds[wave * 2 * ROWS_PER_WAVE * E];
    const int eoff = gidx * E + s * EPL;   // this lane's 8 contiguous elements

    // work in y = flat * (10 * log2(e)): then the mask update is
    // y += log2(m)*10 and softmax is exp2(y - max y)  (native v_log/v_exp)
    const float CO  = 14.4269504088896340736f;   // 10 / (TAU... ) * log2(e)
    const float EPS = 1.1754943508222875e-38f;   // np.finfo(f32).tiny

    float y[EPL];
#pragma unroll
    for (int j = 0; j < EPL; ++j)
        y[j] = (ws[eoff + j] + ws[ROWS_PER_WAVE * E + eoff + j]) * CO;

    float kh[EPL], oh[EPL];
#pragma unroll
    for (int j = 0; j < EPL; ++j) { kh[j] = 0.0f; oh[j] = 0.0f; }

#pragma unroll
    for (int it = 0; it < KSEL; ++it) {
        if (it > 0) {                  // iter 0: onehot==0 -> log(1)==0, skip
#pragma unroll
            for (int j = 0; j < EPL; ++j) {
                float m = fmaxf(1.0f - oh[j], EPS);
                y[j] = fmaf(__builtin_amdgcn_logf(m), 10.0f, y[j]);
            }
        }
        float mx = y[0];
#pragma unroll
        for (int j = 1; j < EPL; ++j) mx = fmaxf(mx, y[j]);
        mx = group8_max(mx);

        float e[EPL], sum = 0.0f;
#pragma unroll
        for (int j = 0; j < EPL; ++j) {
            e[j] = __builtin_amdgcn_exp2f(y[j] - mx);
            sum += e[j];
        }
        sum = group8_sum(sum);

        float inv = __builtin_amdgcn_rcpf(sum);
#pragma unroll
        for (int j = 0; j < EPL; ++j) {
            oh[j] = e[j] * inv;
            kh[j] += oh[j];
        }
    }

    // ---- hard top-8 per row: iterative argmax, lowest-index tie-break
    float sel[EPL];
    unsigned hm = 0;
#pragma unroll
    for (int j = 0; j < EPL; ++j) sel[j] = kh[j];
    const float NEG_INF = -__builtin_inff();

#pragma unroll
    for (int t = 0; t < KSEL; ++t) {
        float v = sel[0];
        int   idx = s * EPL;
#pragma unroll
        for (int j = 1; j < EPL; ++j)
            if (sel[j] > v) { v = sel[j]; idx = s * EPL + j; }  // ties keep lower j
        argmax_step<DPP_XOR1>(v, idx);
        argmax_step<DPP_XOR2>(v, idx);
        argmax_step<DPP_HMIR>(v, idx);
#pragma unroll
        for (int j = 0; j < EPL; ++j)
            if (idx == s * EPL + j) { hm |= 1u << j; sel[j] = NEG_INF; }
    }

    // straight-through estimator, reference association order
    float r[EPL];
#pragma unroll
    for (int j = 0; j < EPL; ++j) {
        float hard = (float)((hm >> j) & 1u);
        r[j] = (hard - kh[j]) + kh[j];
    }

    // write-once outputs: non-temporal 128-bit stores, skip L2 residency
    float* mp = out + row * E + s * EPL;
    float* kp = out + (unsigned long long)nrows * E + row * E + s * EPL;
    v4f r_lo = {r[0], r[1], r[2], r[3]};
    v4f r_hi = {r[4], r[5], r[6], r[7]};
    v4f k_lo = {kh[0], kh[1], kh[2], kh[3]};
    v4f k_hi = {kh[4], kh[5], kh[6], kh[7]};
    __builtin_nontemporal_store(r_lo, (v4f*)mp);
    __builtin_nontemporal_store(r_hi, (v4f*)(mp + 4));
    __builtin_nontemporal_store(k_lo, (v4f*)kp);
    __builtin_nontemporal_store(k_hi, (v4f*)(kp + 4));
}

extern "C" void kernel_launch(void* const* d_in, const int* in_sizes, int n_in,
                              void* d_out, int out_size, void* d_ws, size_t ws_size,
                              hipStream_t stream) {
    const float* scores = (const float*)d_in[0];  // [32, 16384, 64] f32
    const float* g      = (const float*)d_in[1];  // [32*16384, 64] f32
    float* out = (float*)d_out;                   // [mask | khot], each nrows*64

    int nrows = in_sizes[0] / E;                  // 524288
    int blocks = (nrows + ROWS_PER_BLOCK - 1) / ROWS_PER_BLOCK;
    gumbel_relaxed_topk_kernel<<<blocks, 256, 0, stream>>>(scores, g, out, nrows);
}